// MamboPoolingWithClusterAttention_43104291783365
// MI455X (gfx1250) — compile-verified
//
#include <hip/hip_runtime.h>
#include <hip/hip_bf16.h>

typedef __attribute__((ext_vector_type(2))) float v2f;
typedef __attribute__((ext_vector_type(4))) float f4;
typedef __attribute__((ext_vector_type(8))) float v8f;

#define NODES   200000
#define EMBED   256
#define NGRAPH  256
#define CHUNK   128

// ---------------- zero workspace accumulators ----------------
__global__ __launch_bounds__(256) void zero_kernel(float* __restrict__ p, int n) {
    int i = blockIdx.x * 256 + threadIdx.x;
    if (i < n) p[i] = 0.0f;
}

// ---------------- segment sum over sorted batch ----------------
// block = 256 threads (thread == embedding column), one block per CHUNK nodes.
// batch is sorted, so accumulate runs in a register and flush on graph change.
__global__ __launch_bounds__(256) void segsum_kernel(const float* __restrict__ x,
                                                     const int*   __restrict__ batch,
                                                     float* __restrict__ sums,    // [256][256]
                                                     float* __restrict__ counts)  // [256]
{
    const int c  = threadIdx.x;
    int i0 = blockIdx.x * CHUNK;
    int i1 = i0 + CHUNK; if (i1 > NODES) i1 = NODES;
    if (i0 >= i1) return;

    int   cur = batch[i0];
    float acc = 0.0f;
    int   run = 0;
    for (int i = i0; i < i1; ++i) {
        int g = batch[i];                 // uniform across the block
        if (g != cur) {                   // uniform branch
            atomicAdd(&sums[cur * EMBED + c], acc);
            if (c == 0) atomicAdd(&counts[cur], (float)run);
            cur = g; acc = 0.0f; run = 0;
        }
        acc += __builtin_nontemporal_load(&x[(size_t)i * EMBED + c]);
        ++run;
    }
    atomicAdd(&sums[cur * EMBED + c], acc);
    if (c == 0) atomicAdd(&counts[cur], (float)run);
}

// ---------------- mean: g = sums / max(counts,1) ----------------
__global__ __launch_bounds__(256) void mean_kernel(const float* __restrict__ sums,
                                                   const float* __restrict__ counts,
                                                   float* __restrict__ g)
{
    int m = blockIdx.x, c = threadIdx.x;
    float cnt = counts[m];
    float inv = 1.0f / (cnt > 1.0f ? cnt : 1.0f);
    g[m * EMBED + c] = sums[m * EMBED + c] * inv;
}

// ---------------- D = A @ W^T + bias, all 256x256, via V_WMMA_F32_16X16X4_F32 ----
// One wave (32 lanes) per 16x16 output tile; K loop of 64 WMMAs.
// A-frag (16x4 f32): lane 0-15 row M=lane, VGPR0=K+2*half, VGPR1=K+2*half+1.
// B-frag (4x16):     lane holds column N=lane&15; same K selection; B[k][n]=W[n][k].
// C/D: VGPR r -> D[m0 + r + 8*half][n0 + (lane&15)].
__global__ __launch_bounds__(32) void gemm256_wmma(const float* __restrict__ A,
                                                   const float* __restrict__ W,
                                                   const float* __restrict__ bias,
                                                   float* __restrict__ D)
{
    const int lane = threadIdx.x;
    const int half = lane >> 4;
    const int lr   = lane & 15;
    const int m0   = blockIdx.x * 16;
    const int n0   = blockIdx.y * 16;

    const float* __restrict__ arow = A + (size_t)(m0 + lr) * EMBED;
    const float* __restrict__ wrow = W + (size_t)(n0 + lr) * EMBED;

    v8f c;
    const float b0 = bias[n0 + lr];
#pragma unroll
    for (int r = 0; r < 8; ++r) c[r] = b0;

#pragma unroll 8
    for (int k = 0; k < EMBED; k += 4) {
        v2f a, b;
        a.x = arow[k + 2 * half + 0];
        a.y = arow[k + 2 * half + 1];
        b.x = wrow[k + 2 * half + 0];
        b.y = wrow[k + 2 * half + 1];
        c = __builtin_amdgcn_wmma_f32_16x16x4_f32(
                /*neg_a=*/false, a, /*neg_b=*/false, b,
                /*c_mod=*/(short)0, c, /*reuse_a=*/false, /*reuse_b=*/false);
    }

#pragma unroll
    for (int r = 0; r < 8; ++r)
        D[(size_t)(m0 + r + 8 * half) * EMBED + (n0 + lr)] = c[r];
}

// ---------------- gather/broadcast: out[i] = out_graph[batch[i]] ----------------
__global__ __launch_bounds__(256) void gather_kernel(const f4* __restrict__ graph4, // [256][64]
                                                     const int* __restrict__ batch,
                                                     f4* __restrict__ out4)         // [NODES][64]
{
    long long t = (long long)blockIdx.x * 256 + threadIdx.x;
    int node = (int)(t >> 6);
    int c4   = (int)(t & 63);
    if (node >= NODES) return;
    int g = batch[node];
    f4 val = graph4[g * 64 + c4];
    __builtin_nontemporal_store(val, &out4[(size_t)node * 64 + c4]);
}

extern "C" void kernel_launch(void* const* d_in, const int* in_sizes, int n_in,
                              void* d_out, int out_size, void* d_ws, size_t ws_size,
                              hipStream_t stream)
{
    const float* x          = (const float*)d_in[0];
    const float* in_proj_w  = (const float*)d_in[1];
    const float* in_proj_b  = (const float*)d_in[2];
    const float* out_proj_w = (const float*)d_in[3];
    const float* out_proj_b = (const float*)d_in[4];
    const int*   batch      = (const int*)d_in[5];

    const float* Wv = in_proj_w + 2 * EMBED * EMBED;   // rows [2E, 3E)
    const float* bv = in_proj_b + 2 * EMBED;

    float* ws        = (float*)d_ws;
    float* sums      = ws;                      // 256*256
    float* counts    = sums + EMBED * NGRAPH;   // 256
    float* g         = counts + NGRAPH;         // 256*256
    float* v         = g + EMBED * NGRAPH;      // 256*256
    float* out_graph = v + EMBED * NGRAPH;      // 256*256

    // 1. zero accumulators (sums + counts)
    {
        int n = EMBED * NGRAPH + NGRAPH;
        zero_kernel<<<(n + 255) / 256, 256, 0, stream>>>(sums, n);
    }
    // 2. segment sum
    {
        int nblk = (NODES + CHUNK - 1) / CHUNK;
        segsum_kernel<<<nblk, 256, 0, stream>>>(x, batch, sums, counts);
    }
    // 3. mean
    mean_kernel<<<NGRAPH, EMBED, 0, stream>>>(sums, counts, g);

    // 4. v = g @ Wv^T + bv ; out_graph = v @ Wout^T + bout
    dim3 tiles(EMBED / 16, EMBED / 16);
    gemm256_wmma<<<tiles, 32, 0, stream>>>(g, Wv, bv, v);
    gemm256_wmma<<<tiles, 32, 0, stream>>>(v, out_proj_w, out_proj_b, out_graph);

    // 5. broadcast back to nodes
    {
        long long threads = (long long)NODES * (EMBED / 4);
        int nblk = (int)((threads + 255) / 256);
        gather_kernel<<<nblk, 256, 0, stream>>>((const f4*)out_graph, batch, (f4*)d_out);
    }
}